// LSTM_14688788152407
// MI455X (gfx1250) — compile-verified
//
#include <hip/hip_runtime.h>
#include <math.h>

// ----------------------------------------------------------------------------
// 8-layer stacked LSTM, batch=1, seq=1, H=2048, fp32.
// HBM-bound GEMV workload (AI ~0.25 FLOP/B). Strategy:
//   1) hh_gemv: all 8 recurrent GEMVs (h0[l] @ w_hh[l].T) in parallel
//      (16384 waves, streams 537 MB) -> deterministic k-split partials.
//   2) per layer l (serial dependency only through w_ih):
//        ih_gemv: out_{l-1} @ w_ih[l].T (2048 waves, 67 MB)
//        act:     gate sum + biases + sigmoid/tanh -> next layer input
// GEMV accumulation runs on V_WMMA_F32_16X16X4_F32 (compute is free at this
// arithmetic intensity; the wave loads a 16x4 weight tile = 256B per WMMA).
// ----------------------------------------------------------------------------

#define HDIM   2048
#define H4     8192           // 4*HDIM (gates per layer)
#define NLAYER 8
#define KSPLIT 4
#define KS     (HDIM / KSPLIT)  // 512 k per wave

typedef float v2f __attribute__((ext_vector_type(2)));
typedef float v8f __attribute__((ext_vector_type(8)));

// y[rowbase .. rowbase+15] partial = W[rowbase..+15][kbase..kbase+KS) . v[..]
// WMMA mapping: D(16x16) = A(16x4) * B(4x16) + C
//   A = weight tile. ISA 32-bit A 16x4 layout: lane L holds
//       W[m = L%16][k0 + 2*(L/16)] and [.. + 1]  -> one b64 load per lane.
//   B = v broadcast across all 16 columns. Assumed B lane mapping mirrors A:
//       VGPR0 = {K0 (lanes 0-15), K2 (lanes 16-31)}, VGPR1 = {K1, K3}.
//       Broadcast makes every D column identical -> extract column 0.
__device__ __forceinline__ void gemv16_wmma(const float* __restrict__ Wrow, // W + rowbase*HDIM
                                            const float* __restrict__ v,
                                            float* __restrict__ out,        // partial + rowbase
                                            int kbase, int lane)
{
    const int m     = lane & 15;
    const int halfk = (lane >> 4) << 1;          // 0 or 2
    const float* wp = Wrow + (size_t)m * HDIM + kbase + halfk;
    const float* vp = v + kbase + halfk;

    v8f acc = {0.f, 0.f, 0.f, 0.f, 0.f, 0.f, 0.f, 0.f};

    for (int k0 = 0; k0 < KS; k0 += 32) {
        // stream-ahead hint on the weight path (gfx1250 global_prefetch_b8)
        __builtin_prefetch(wp + k0 + 64, 0, 1);
        #pragma unroll
        for (int kk = 0; kk < 32; kk += 4) {
            v2f a = *(const v2f*)(wp + k0 + kk);   // 8B-aligned (even float idx)
            v2f b = *(const v2f*)(vp + k0 + kk);
            acc = __builtin_amdgcn_wmma_f32_16x16x4_f32(
                false, a, false, b, (short)0, acc, false, false);
        }
    }

    // D layout: lane L holds column N=L%16, rows (L<16 ? 0-7 : 8-15) in acc[0..7].
    // All columns identical -> lanes 0 and 16 write the 16 row sums.
    if ((lane & 15) == 0) {
        const int mb = (lane >> 4) * 8;
        #pragma unroll
        for (int r = 0; r < 8; ++r) out[mb + r] = acc[r];
    }
}

// All 8 layers' recurrent GEMVs in parallel: hh_part[l][s][j] partials.
// 16384 waves (2048 blocks x 256 threads).
__global__ void lstm_hh_gemv(const float* __restrict__ w_hh,
                             const float* __restrict__ h0,
                             float* __restrict__ hh_part)
{
    const int wave = (int)((blockIdx.x * blockDim.x + threadIdx.x) >> 5);
    const int lane = threadIdx.x & 31;
    const int l  = wave >> 11;        // / 2048
    const int rm = wave & 2047;
    const int s  = rm >> 9;           // k-split 0..3
    const int rg = rm & 511;          // row group 0..511 (16 rows each)

    const float* W = w_hh + (size_t)l * H4 * HDIM + (size_t)rg * 16 * HDIM;
    const float* v = h0 + (size_t)l * HDIM;
    float* out = hh_part + ((size_t)(l * KSPLIT + s)) * H4 + rg * 16;
    gemv16_wmma(W, v, out, s * KS, lane);
}

// One layer's input GEMV: ih_part[s][j] partials. 2048 waves (256 blocks).
__global__ void lstm_ih_gemv(const float* __restrict__ Wl,   // w_ih + l*H4*HDIM
                             const float* __restrict__ vin,  // layer input [HDIM]
                             float* __restrict__ ih_part)
{
    const int wave = (int)((blockIdx.x * blockDim.x + threadIdx.x) >> 5);
    const int lane = threadIdx.x & 31;
    const int s  = wave >> 9;
    const int rg = wave & 511;

    const float* W = Wl + (size_t)rg * 16 * HDIM;
    float* out = ih_part + (size_t)s * H4 + rg * 16;
    gemv16_wmma(W, vin, out, s * KS, lane);
}

__device__ __forceinline__ float sigm(float x) { return 1.0f / (1.0f + expf(-x)); }

// Per-layer gate combine + activations. 2048 threads.
__global__ void lstm_act(const float* __restrict__ b_ih,
                         const float* __restrict__ b_hh,
                         const float* __restrict__ c0,
                         const float* __restrict__ hh_part,
                         const float* __restrict__ ih_part,
                         float* __restrict__ vec_out,   // input to layer l+1
                         float* __restrict__ dout,
                         int l)
{
    const int j = (int)(blockIdx.x * blockDim.x + threadIdx.x);
    if (j >= HDIM) return;

    float gate[4];
    #pragma unroll
    for (int g = 0; g < 4; ++g) {
        const int idx = g * HDIM + j;
        float t = b_ih[(size_t)l * H4 + idx] + b_hh[(size_t)l * H4 + idx];
        #pragma unroll
        for (int s = 0; s < KSPLIT; ++s)
            t += hh_part[((size_t)(l * KSPLIT + s)) * H4 + idx]
               + ih_part[(size_t)s * H4 + idx];
        gate[g] = t;
    }
    // PyTorch gate order: i, f, g, o
    const float ig = sigm(gate[0]);
    const float fg = sigm(gate[1]);
    const float gg = tanhf(gate[2]);
    const float og = sigm(gate[3]);
    const float cn = fg * c0[(size_t)l * HDIM + j] + ig * gg;
    const float hn = og * tanhf(cn);

    if (l < NLAYER - 1) vec_out[j] = tanhf(hn);  // inter-layer tanh
    else                dout[j]    = hn;         // final layer: no extra tanh
}

extern "C" void kernel_launch(void* const* d_in, const int* in_sizes, int n_in,
                              void* d_out, int out_size, void* d_ws, size_t ws_size,
                              hipStream_t stream) {
    (void)in_sizes; (void)n_in; (void)out_size; (void)ws_size;

    const float* x    = (const float*)d_in[0];   // [2048]
    const float* w_ih = (const float*)d_in[1];   // [8, 8192, 2048]
    const float* w_hh = (const float*)d_in[2];   // [8, 8192, 2048]
    const float* b_ih = (const float*)d_in[3];   // [8, 8192]
    const float* b_hh = (const float*)d_in[4];   // [8, 8192]
    const float* h0   = (const float*)d_in[5];   // [8, 1, 2048]
    const float* c0   = (const float*)d_in[6];   // [8, 1, 2048]
    float* dout = (float*)d_out;                 // [2048]

    // Workspace layout (floats):
    //   hh_part : [8][4][8192]  = 262144
    //   ih_part : [4][8192]     =  32768  (reused each layer; stream-ordered)
    //   vec     : [8][2048]     =  16384  (vec[l] = output of layer l)
    float* ws      = (float*)d_ws;
    float* hh_part = ws;
    float* ih_part = ws + (size_t)NLAYER * KSPLIT * H4;
    float* vec     = ih_part + (size_t)KSPLIT * H4;

    // 1) All recurrent GEMVs in parallel: 16384 waves.
    lstm_hh_gemv<<<2048, 256, 0, stream>>>(w_hh, h0, hh_part);

    // 2) Serial chain of input GEMVs + gate activations.
    for (int l = 0; l < NLAYER; ++l) {
        const float* vin = (l == 0) ? x : (vec + (size_t)(l - 1) * HDIM);
        lstm_ih_gemv<<<256, 256, 0, stream>>>(w_ih + (size_t)l * H4 * HDIM, vin, ih_part);
        lstm_act<<<HDIM / 256, 256, 0, stream>>>(b_ih, b_hh, c0, hh_part, ih_part,
                                                 vec + (size_t)l * HDIM, dout, l);
    }
}